// BiasBlock_69037304316258
// MI455X (gfx1250) — compile-verified
//
#include <hip/hip_runtime.h>
#include <math.h>

#define C_DIM 128
#define WDIM  512
#define SDIM  2560
#define RANK  10
#define INV_SQRT_RANK 0.31622776601683794f  // 1/sqrt(10)

typedef float v2f __attribute__((ext_vector_type(2)));
typedef float v8f __attribute__((ext_vector_type(8)));

// ---------------------------------------------------------------------------
// Kernel 1: style -> modulated + demodulated weight, written TRANSPOSED
// Wt[k][n] = Wm_normalized[n][k]  (so GEMM B-fragments read contiguous rows)
// Single block of 256 threads; work is ~1.3M MACs -> negligible.
// ---------------------------------------------------------------------------
__global__ __launch_bounds__(256)
void modweights_kernel(const float* __restrict__ w,      // [512]
                       const float* __restrict__ affW,   // [2560,512]
                       const float* __restrict__ affb,   // [2560]
                       const float* __restrict__ W,      // [128,128]
                       float* __restrict__ WtOut)        // [128,128] transposed
{
    __shared__ float s_w[WDIM];
    __shared__ float s_styles[SDIM];
    __shared__ float s_norm[C_DIM];

    const int tid = threadIdx.x;
    for (int i = tid; i < WDIM; i += 256) s_w[i] = w[i];
    if (tid < C_DIM) s_norm[tid] = 0.0f;
    __syncthreads();

    // styles = affW @ w + affb
    for (int s = tid; s < SDIM; s += 256) {
        const float* row = affW + (size_t)s * WDIM;
        float acc = affb[s];
        for (int k = 0; k < WDIM; ++k) acc += row[k] * s_w[k];
        s_styles[s] = acc;
    }
    __syncthreads();

    // pass 1: accumulate per-output-row squared norms of Wm
    for (int idx = tid; idx < C_DIM * C_DIM; idx += 256) {
        const int co = idx >> 7, ci = idx & 127;
        float mod = 0.0f;
        #pragma unroll
        for (int r = 0; r < RANK; ++r)
            mod += s_styles[co * RANK + r] * s_styles[C_DIM * RANK + r * C_DIM + ci];
        const float wm = W[idx] * (mod * INV_SQRT_RANK + 1.0f);
        atomicAdd(&s_norm[co], wm * wm);
    }
    __syncthreads();

    if (tid < C_DIM) s_norm[tid] = 1.0f / (sqrtf(s_norm[tid]) + 1e-8f);
    __syncthreads();

    // pass 2: recompute Wm, scale, store transposed
    for (int idx = tid; idx < C_DIM * C_DIM; idx += 256) {
        const int co = idx >> 7, ci = idx & 127;
        float mod = 0.0f;
        #pragma unroll
        for (int r = 0; r < RANK; ++r)
            mod += s_styles[co * RANK + r] * s_styles[C_DIM * RANK + r * C_DIM + ci];
        const float wm = W[idx] * (mod * INV_SQRT_RANK + 1.0f);
        WtOut[ci * C_DIM + co] = wm * s_norm[co];
    }
}

// ---------------------------------------------------------------------------
// Kernel 2: agg[n][:] = el_b  (float4 broadcast)
// ---------------------------------------------------------------------------
__global__ void agg_init_kernel(float* __restrict__ agg,
                                const float* __restrict__ bias, int n_rows)
{
    const int i = blockIdx.x * blockDim.x + threadIdx.x;   // float4 index
    const int total = n_rows * (C_DIM / 4);
    if (i < total) {
        const float4* b4 = (const float4*)bias;
        ((float4*)agg)[i] = b4[i & (C_DIM / 4 - 1)];
    }
}

// ---------------------------------------------------------------------------
// Kernel 3: segment-sum scatter: agg[dst] += el_W[src]; one wave32 per edge,
// 4 channels per lane -> coalesced 128B loads + global_atomic_add_f32.
// ---------------------------------------------------------------------------
__global__ __launch_bounds__(256)
void edge_scatter_kernel(const int* __restrict__ src, const int* __restrict__ dst,
                         const float* __restrict__ elW, float* __restrict__ agg, int E)
{
    const int wid  = (blockIdx.x * blockDim.x + threadIdx.x) >> 5;
    const int lane = threadIdx.x & 31;
    if (wid >= E) return;
    const float* wrow = elW + (size_t)src[wid] * C_DIM;
    float*       arow = agg + (size_t)dst[wid] * C_DIM;
    #pragma unroll
    for (int j = 0; j < 4; ++j) {
        const int c = lane + 32 * j;
        atomicAdd(arow + c, wrow[c]);
    }
}

// ---------------------------------------------------------------------------
// Kernel 4: fused SynthesisLayer GEMM on the WMMA pipe.
//   y = act(x @ WmT + bias, slope) + ns*noise       [per 16x128 slab per wave]
//   if ext: y = leaky(ext + y, final_slope)          (fused residual epilogue)
// Weight (transposed, K-major) staged in 64KB LDS; 8 waves x 16 rows = 128
// rows per block. WMMA: V_WMMA_F32_16X16X4_F32, K-loop 32 steps x 8 N-tiles.
// ---------------------------------------------------------------------------
__global__ __launch_bounds__(256)
void synth_wmma_kernel(const float* X,                         // [n_rows,128]
                       const float* __restrict__ Wt,           // [128,128] K-major
                       const float* __restrict__ bias,         // [128]
                       const float* __restrict__ ns_p,         // scalar
                       const float* __restrict__ noise,        // [n_rows,128]
                       const float* ext,                       // nullable [n_rows,128]
                       float slope, float final_slope,
                       float* Y, int n_rows)
{
    __shared__ float s_Wt[C_DIM * C_DIM];                      // 64 KB

    const int tid = threadIdx.x;
    {   // stage weight to LDS (float4 copies)
        const float4* s4 = (const float4*)Wt;
        float4*       d4 = (float4*)s_Wt;
        for (int i = tid; i < C_DIM * C_DIM / 4; i += 256) d4[i] = s4[i];
    }
    __syncthreads();

    const int wave = tid >> 5;
    const int lane = tid & 31;
    const int hi   = lane >> 4;       // 0: lanes 0-15, 1: lanes 16-31
    const int lo   = lane & 15;
    const int r0   = blockIdx.x * 128 + wave * 16;

    int arow = r0 + lo;                         // A-fragment row for this lane
    if (arow >= n_rows) arow = n_rows - 1;      // clamp: EXEC must stay all-1s
    const float* xrow = X + (size_t)arow * C_DIM;

    v8f acc[8] = {};                             // 8 x (16x16) f32 tiles

    for (int k0 = 0; k0 < C_DIM; k0 += 4) {
        v2f a;                                   // A 16x4: K = k0 + 2*hi + v
        a.x = xrow[k0 + 2 * hi];
        a.y = xrow[k0 + 2 * hi + 1];
        #pragma unroll
        for (int t = 0; t < 8; ++t) {
            const int n = t * 16 + lo;           // B 4x16: N in lanes, K per VGPR
            v2f b;
            b.x = s_Wt[(k0 + 2 * hi)     * C_DIM + n];
            b.y = s_Wt[(k0 + 2 * hi + 1) * C_DIM + n];
            acc[t] = __builtin_amdgcn_wmma_f32_16x16x4_f32(
                         false, a, false, b, (short)0, acc[t], false, false);
        }
    }

    const float ns = ns_p[0];
    #pragma unroll
    for (int t = 0; t < 8; ++t) {
        const int n  = t * 16 + lo;
        const float bn = bias[n];
        #pragma unroll
        for (int r = 0; r < 8; ++r) {            // D layout: row = r + 8*hi
            const int row = r0 + r + 8 * hi;
            if (row < n_rows) {
                float v = acc[t][r] + bn;
                v = (v >= 0.0f) ? v : slope * v;
                v += ns * noise[(size_t)row * C_DIM + n];
                if (ext) {
                    v += ext[(size_t)row * C_DIM + n];
                    v = (v >= 0.0f) ? v : final_slope * v;
                }
                Y[(size_t)row * C_DIM + n] = v;
            }
        }
    }
}

// ---------------------------------------------------------------------------
extern "C" void kernel_launch(void* const* d_in, const int* in_sizes, int n_in,
                              void* d_out, int out_size, void* d_ws, size_t ws_size,
                              hipStream_t stream)
{
    const float* x        = (const float*)d_in[0];
    const float* w        = (const float*)d_in[1];
    const int*   ei       = (const int*)  d_in[2];
    const float* l1_affW  = (const float*)d_in[3];
    const float* l1_affb  = (const float*)d_in[4];
    const float* l1_W     = (const float*)d_in[5];
    const float* l1_b     = (const float*)d_in[6];
    const float* l1_ns    = (const float*)d_in[7];
    const float* l1_noise = (const float*)d_in[8];
    const float* l2_affW  = (const float*)d_in[9];
    const float* l2_affb  = (const float*)d_in[10];
    const float* l2_W     = (const float*)d_in[11];
    const float* l2_b     = (const float*)d_in[12];
    const float* l2_ns    = (const float*)d_in[13];
    const float* l2_noise = (const float*)d_in[14];
    const float* e2_affW  = (const float*)d_in[15];
    const float* e2_affb  = (const float*)d_in[16];
    const float* e2_W     = (const float*)d_in[17];
    const float* e2_b     = (const float*)d_in[18];
    const float* e2_ns    = (const float*)d_in[19];
    const float* e2_noise = (const float*)d_in[20];
    const float* el_W     = (const float*)d_in[21];
    const float* el_b     = (const float*)d_in[22];

    const int N = in_sizes[0] / C_DIM;
    const int E = in_sizes[2] / 2;

    // workspace layout
    float* Wt1 = (float*)d_ws;
    float* Wt2 = Wt1 + C_DIM * C_DIM;
    float* Wt3 = Wt2 + C_DIM * C_DIM;
    float* h   = Wt3 + C_DIM * C_DIM;            // [N,128]
    float* agg = h + (size_t)N * C_DIM;          // [N,128]

    // 1) modulated + demodulated weights (tiny)
    modweights_kernel<<<1, 256, 0, stream>>>(w, l1_affW, l1_affb, l1_W, Wt1);
    modweights_kernel<<<1, 256, 0, stream>>>(w, l2_affW, l2_affb, l2_W, Wt2);
    modweights_kernel<<<1, 256, 0, stream>>>(w, e2_affW, e2_affb, e2_W, Wt3);

    // 2) edge branch aggregation
    const int initN = N * (C_DIM / 4);
    agg_init_kernel<<<(initN + 255) / 256, 256, 0, stream>>>(agg, el_b, N);
    const long long ethreads = (long long)E * 32;
    edge_scatter_kernel<<<(int)((ethreads + 255) / 256), 256, 0, stream>>>(
        ei, ei + E, el_W, agg, E);

    // 3) three fused WMMA synth layers
    const int gblocks = (N + 127) / 128;
    synth_wmma_kernel<<<gblocks, 256, 0, stream>>>(
        x, Wt1, l1_b, l1_ns, l1_noise, nullptr, 0.01f, 1.0f, h, N);
    synth_wmma_kernel<<<gblocks, 256, 0, stream>>>(      // in-place: safe per-wave
        h, Wt2, l2_b, l2_ns, l2_noise, nullptr, 1.0f, 1.0f, h, N);
    synth_wmma_kernel<<<gblocks, 256, 0, stream>>>(      // + fused h+agg2, final leaky
        agg, Wt3, e2_b, e2_ns, e2_noise, h, 1.0f, 0.01f, (float*)d_out, N);

    (void)n_in; (void)out_size; (void)ws_size;
}